// GraphormerAttentionHead_3289944949217
// MI455X (gfx1250) — compile-verified
//
#include <hip/hip_runtime.h>
#include <hip/hip_bf16.h>
#include <math.h>

typedef __attribute__((ext_vector_type(2))) float v2f;
typedef __attribute__((ext_vector_type(8))) float v8f;

#define N_NODES 1024
#define DIM     256
#define GSZ     64
#define NGRAPH  16
#define EDIM    16
#define LPATH   5

// ---------------------------------------------------------------------------
// Kernel 1: q/k/v = x @ W.T + bias   (one 16x16 tile per wave, f32 WMMA K=4)
// A frag (16x4 f32): lane m=lane&15; VGPR0/1 hold K = k0+2*half {+0,+1}
// B frag (4x16 f32): lane n=lane&15; VGPR0/1 hold K = k0+2*half {+0,+1}
// C/D (16x16 f32):   VGPR r -> row = r + 8*half, col = lane&15
// ---------------------------------------------------------------------------
__global__ __launch_bounds__(32)
void qkv_gemm(const float* __restrict__ X,
              const float* __restrict__ Wq, const float* __restrict__ bq,
              const float* __restrict__ Wk, const float* __restrict__ bk,
              const float* __restrict__ Wv, const float* __restrict__ bv,
              float* __restrict__ q, float* __restrict__ k, float* __restrict__ v) {
  const float* W; const float* bias; float* Out;
  if (blockIdx.z == 0)      { W = Wq; bias = bq; Out = q; }
  else if (blockIdx.z == 1) { W = Wk; bias = bk; Out = k; }
  else                      { W = Wv; bias = bv; Out = v; }

  const int lane = threadIdx.x & 31;
  const int mn   = lane & 15;
  const int kk   = (lane >> 4) << 1;        // 0 or 2
  const int row0 = blockIdx.x * 16;
  const int col0 = blockIdx.y * 16;

  const float* arow = X + (size_t)(row0 + mn) * DIM;   // A: rows of x
  const float* brow = W + (size_t)(col0 + mn) * DIM;   // B[k][n] = W[n][k]

  v8f acc = {};
  for (int k0 = 0; k0 < DIM; k0 += 4) {
    v2f a, b;
    a.x = arow[k0 + kk];  a.y = arow[k0 + kk + 1];
    b.x = brow[k0 + kk];  b.y = brow[k0 + kk + 1];
    acc = __builtin_amdgcn_wmma_f32_16x16x4_f32(false, a, false, b,
                                                (short)0, acc, false, false);
  }
  const float bb    = bias[col0 + mn];
  const int   rbase = row0 + ((lane >> 4) << 3);
#pragma unroll
  for (int r = 0; r < 8; ++r)
    Out[(size_t)(rbase + r) * DIM + col0 + mn] = acc[r] + bb;
}

// ---------------------------------------------------------------------------
// Kernel 2: block-diagonal scores  S[i][jl] = (q_i . k_j) * 256^-0.5
// One 16x16 tile per wave; grid (rowTile=4, colTile=4, graph=16)
// ---------------------------------------------------------------------------
__global__ __launch_bounds__(32)
void scores_gemm(const float* __restrict__ q, const float* __restrict__ kmat,
                 float* __restrict__ Sblk) {
  const int lane = threadIdx.x & 31;
  const int mn   = lane & 15;
  const int kk   = (lane >> 4) << 1;
  const int g    = blockIdx.z;
  const int row0 = g * GSZ + blockIdx.x * 16;   // global row
  const int colg = blockIdx.y * 16;             // column inside graph

  const float* arow = q    + (size_t)(row0 + mn) * DIM;
  const float* brow = kmat + (size_t)(g * GSZ + colg + mn) * DIM; // B[k][n]=k[j][k]

  v8f acc = {};
  for (int k0 = 0; k0 < DIM; k0 += 4) {
    v2f a, b;
    a.x = arow[k0 + kk];  a.y = arow[k0 + kk + 1];
    b.x = brow[k0 + kk];  b.y = brow[k0 + kk + 1];
    acc = __builtin_amdgcn_wmma_f32_16x16x4_f32(false, a, false, b,
                                                (short)0, acc, false, false);
  }
  const float scale = 0.0625f;                  // 256^-0.5
  const int   rl    = blockIdx.x * 16 + ((lane >> 4) << 3); // local row in graph
#pragma unroll
  for (int r = 0; r < 8; ++r)
    Sblk[(size_t)(g * GSZ + rl + r) * GSZ + colg + mn] = acc[r] * scale;
}

// ---------------------------------------------------------------------------
// Kernel 3: edge encoding + full-row stable softmax, one row per workgroup.
// Builds the full 1024-entry logit row in LDS (on-block: S+b+c, off-block:
// (b+c)*-1e6), reduces max and sum(exp), writes normalized block-diag probs
// in place over Sblk.
// ---------------------------------------------------------------------------
__global__ __launch_bounds__(256)
void edge_softmax(const float* __restrict__ edge_attr,
                  const float* __restrict__ Bmat,
                  const int*   __restrict__ edge_paths,
                  const int*   __restrict__ ptr,
                  const float* __restrict__ edge_vec,
                  float*       __restrict__ SPblk) {
  __shared__ float lg[N_NODES];
  __shared__ float red[256];
  __shared__ float ev[LPATH * EDIM];

  const int tid = threadIdx.x;
  const int i   = blockIdx.x;
  if (tid < LPATH * EDIM) ev[tid] = edge_vec[tid];
  __syncthreads();

  // gid = searchsorted(ptr, i, 'right') - 1
  int g = 0;
  for (int t = 1; t <= NGRAPH; ++t)
    if (ptr[t] <= i) g = t;
  const int g0 = ptr[g];
  const int g1 = ptr[g + 1];

  for (int j = tid; j < N_NODES; j += 256) {
    if (j + 256 < N_NODES)
      __builtin_prefetch(edge_paths + ((size_t)i * N_NODES + j + 256) * LPATH, 0, 0);
    const int* ep = edge_paths + ((size_t)i * N_NODES + j) * LPATH;
    float acc = 0.f;
    int   lens = 0;
#pragma unroll
    for (int l = 0; l < LPATH; ++l) {
      const int e = ep[l];
      if (e >= 0) {
        ++lens;
        const float* ea = edge_attr + (size_t)e * EDIM;
        float dot = 0.f;
        bool  anyne = false;
#pragma unroll
        for (int t = 0; t < EDIM; ++t) {
          const float a = ea[t];
          anyne = anyne || (a != -1.0f);
          dot += a * ev[l * EDIM + t];
        }
        if (anyne) acc += dot;
      }
    }
    const float c   = (lens != 0) ? acc / (float)lens : acc;
    const float bij = Bmat[(size_t)i * N_NODES + j];
    float lv;
    if (j >= g0 && j < g1)
      lv = SPblk[(size_t)i * GSZ + (j - g0)] + bij + c;   // on-block: a+b+c
    else
      lv = (bij + c) * -1.0e6f;                           // off-block: (b+c)*mask_neg
    lg[j] = lv;
  }
  __syncthreads();

  // row max
  float m = -INFINITY;
  for (int j = tid; j < N_NODES; j += 256) m = fmaxf(m, lg[j]);
  red[tid] = m; __syncthreads();
  for (int s = 128; s > 0; s >>= 1) {
    if (tid < s) red[tid] = fmaxf(red[tid], red[tid + s]);
    __syncthreads();
  }
  const float M = red[0];
  __syncthreads();

  // sum(exp)
  float ssum = 0.f;
  for (int j = tid; j < N_NODES; j += 256) ssum += __expf(lg[j] - M);
  red[tid] = ssum; __syncthreads();
  for (int s = 128; s > 0; s >>= 1) {
    if (tid < s) red[tid] += red[tid + s];
    __syncthreads();
  }
  const float inv = 1.0f / red[0];

  // write normalized on-block probabilities (off-block is masked to 0 anyway)
  for (int j = g0 + tid; j < g1; j += 256)
    SPblk[(size_t)i * GSZ + (j - g0)] = __expf(lg[j] - M) * inv;
}

// ---------------------------------------------------------------------------
// Kernel 4: out = P(block-diag 64x64) @ v(64x256), per graph, f32 WMMA
// grid (rowTile=4, colTile=16, graph=16)
// ---------------------------------------------------------------------------
__global__ __launch_bounds__(32)
void out_gemm(const float* __restrict__ Pblk, const float* __restrict__ vmat,
              float* __restrict__ out) {
  const int lane = threadIdx.x & 31;
  const int mn   = lane & 15;
  const int kk   = (lane >> 4) << 1;
  const int g    = blockIdx.z;
  const int rowl = blockIdx.x * 16;     // row inside graph
  const int col0 = blockIdx.y * 16;

  const float* arow = Pblk + (size_t)(g * GSZ + rowl + mn) * GSZ;

  v8f acc = {};
  for (int k0 = 0; k0 < GSZ; k0 += 4) {
    v2f a, b;
    a.x = arow[k0 + kk];  a.y = arow[k0 + kk + 1];
    b.x = vmat[(size_t)(g * GSZ + k0 + kk)     * DIM + col0 + mn];
    b.y = vmat[(size_t)(g * GSZ + k0 + kk + 1) * DIM + col0 + mn];
    acc = __builtin_amdgcn_wmma_f32_16x16x4_f32(false, a, false, b,
                                                (short)0, acc, false, false);
  }
  const int rbase = g * GSZ + rowl + ((lane >> 4) << 3);
#pragma unroll
  for (int r = 0; r < 8; ++r)
    out[(size_t)(rbase + r) * DIM + col0 + mn] = acc[r];
}

// ---------------------------------------------------------------------------
extern "C" void kernel_launch(void* const* d_in, const int* in_sizes, int n_in,
                              void* d_out, int out_size, void* d_ws, size_t ws_size,
                              hipStream_t stream) {
  (void)in_sizes; (void)n_in; (void)out_size; (void)ws_size;

  const float* x          = (const float*)d_in[0];
  const float* edge_attr  = (const float*)d_in[1];
  const float* bmat       = (const float*)d_in[2];
  const int*   edge_paths = (const int*)  d_in[3];
  const int*   ptr        = (const int*)  d_in[4];
  const float* edge_vec   = (const float*)d_in[5];
  const float* Wq = (const float*)d_in[6];
  const float* bq = (const float*)d_in[7];
  const float* Wk = (const float*)d_in[8];
  const float* bk = (const float*)d_in[9];
  const float* Wv = (const float*)d_in[10];
  const float* bv = (const float*)d_in[11];
  float* out = (float*)d_out;

  char*  ws = (char*)d_ws;
  float* q  = (float*)(ws);
  float* k  = (float*)(ws + (size_t)N_NODES * DIM * 4);
  float* v  = (float*)(ws + (size_t)N_NODES * DIM * 8);
  float* SP = (float*)(ws + (size_t)N_NODES * DIM * 12);   // 1024 x 64 f32

  qkv_gemm   <<<dim3(N_NODES / 16, DIM / 16, 3), 32, 0, stream>>>(
      x, Wq, bq, Wk, bk, Wv, bv, q, k, v);
  scores_gemm<<<dim3(GSZ / 16, GSZ / 16, NGRAPH), 32, 0, stream>>>(q, k, SP);
  edge_softmax<<<dim3(N_NODES), 256, 0, stream>>>(
      edge_attr, bmat, edge_paths, ptr, edge_vec, SP);
  out_gemm   <<<dim3(GSZ / 16, DIM / 16, NGRAPH), 32, 0, stream>>>(SP, v, out);
}